// RcnnPostProcess_82884278879271
// MI455X (gfx1250) — compile-verified
//
#include <hip/hip_runtime.h>
#include <hip/hip_bf16.h>
#include <math.h>

#define BB        16
#define NN        2048
#define NC        81        // NUM_CLASSES + 1 (incl. background)
#define TOPK      100
#define NMS_T     0.3f
#define FILT_T    0.1f
#define IMG_WF    1024.0f
#define IMG_HF    1024.0f
#define NTHREADS  1024

// ---- CDNA5 async Global -> LDS DMA (ASYNCcnt path), per ISA 15.18.3 op 98 ----
__device__ __forceinline__ void async_b128_to_lds(const void* gptr, unsigned lds_off) {
  unsigned long long ga = (unsigned long long)(uintptr_t)gptr;
  asm volatile("global_load_async_to_lds_b128 %0, %1, off"
               :: "v"(lds_off), "v"(ga)
               : "memory");
}

__device__ __forceinline__ void wait_asynccnt0() {
#if __has_builtin(__builtin_amdgcn_s_wait_asynccnt)
  __builtin_amdgcn_s_wait_asynccnt(0);
#else
  asm volatile("s_wait_asynccnt 0" ::: "memory");
#endif
}

__device__ __forceinline__ float iou_f(float4 a, float4 b) {
  float aa = (a.z - a.x) * (a.w - a.y);
  float ab = (b.z - b.x) * (b.w - b.y);
  float lx = fmaxf(a.x, b.x), ly = fmaxf(a.y, b.y);
  float rx = fminf(a.z, b.z), ry = fminf(a.w, b.w);
  float iw = fmaxf(rx - lx, 0.0f), ih = fmaxf(ry - ly, 0.0f);
  float inter = iw * ih;
  return inter / (aa + ab - inter + 1e-6f);
}

__global__ __launch_bounds__(NTHREADS) void rcnn_post_kernel(
    const float* __restrict__ boxes,   // [B,N,4]
    const float* __restrict__ scores,  // [B*N,81]
    const float* __restrict__ deltas,  // [B*N,324]
    float* __restrict__ out)           // [B,TOPK,6]
{
  __shared__ __align__(16) float s_dec[NN][4];   // staged raw boxes, then decoded
  __shared__ float         s_key[NN];            // sort key (score or -1)
  __shared__ short         s_val[NN];            // payload: original box index
  __shared__ unsigned char s_cls[NN];            // argmax class

  const int tid = threadIdx.x;
  const int img = blockIdx.x;

  const float* boxB = boxes  + (size_t)img * NN * 4;
  const float* scoB = scores + (size_t)img * NN * NC;
  const float* delB = deltas + (size_t)img * NN * NC * 4;
  float*       outB = out    + (size_t)img * TOPK * 6;

  // ---- Phase 0: kick off async DMA of this image's raw box tile into LDS ----
  #pragma unroll
  for (int e = 0; e < 2; ++e) {
    int i = tid + e * NTHREADS;
    async_b128_to_lds(boxB + (size_t)i * 4, (unsigned)(uintptr_t)&s_dec[i][0]);
  }

  // ---- Phase 1: per-box argmax over 81 classes + gather argmax-class deltas ----
  float4 dd[2];
  float  best[2];
  int    bi[2];
  #pragma unroll
  for (int e = 0; e < 2; ++e) {
    int i = tid + e * NTHREADS;
    const float* sr = scoB + (size_t)i * NC;
    float b0 = sr[0];
    int   c0 = 0;
    #pragma unroll 8
    for (int c = 1; c < NC; ++c) {
      float s = sr[c];
      if (s > b0) { b0 = s; c0 = c; }   // strict >: first max, matches jnp.argmax
    }
    best[e] = b0;
    bi[e]   = c0;
    // (i*324 + c0*4) floats = 16B aligned (1296*i + 16*c0 bytes)
    dd[e] = *(const float4*)(delB + (size_t)i * (NC * 4) + c0 * 4);
  }

  // DMA'd box tile is consumed per-issuing-thread: wave-level wait suffices.
  wait_asynccnt0();

  // ---- Phase 2: decode in place ----
  const float kClip = logf(1000.0f / 16.0f);
  #pragma unroll
  for (int e = 0; e < 2; ++e) {
    int i = tid + e * NTHREADS;
    float4 rb = *(const float4*)&s_dec[i][0];
    // qint16 quantization scale 0.25 (rintf == round-half-even == jnp.round)
    float bx1 = rintf(rb.x * 4.0f) * 0.25f;
    float by1 = rintf(rb.y * 4.0f) * 0.25f;
    float bx2 = rintf(rb.z * 4.0f) * 0.25f;
    float by2 = rintf(rb.w * 4.0f) * 0.25f;
    float w  = bx2 - bx1, h = by2 - by1;
    float cx = bx1 + 0.5f * w, cy = by1 + 0.5f * h;
    float4 d = dd[e];                       // DELTA_STD=1, DELTA_MEAN=0 folded
    float pcx = cx + d.x * w;
    float pcy = cy + d.y * h;
    float pw  = expf(fminf(d.z, kClip)) * w;
    float ph  = expf(fminf(d.w, kClip)) * h;
    float x1 = fminf(fmaxf(pcx - 0.5f * pw, 0.0f), IMG_WF);
    float y1 = fminf(fmaxf(pcy - 0.5f * ph, 0.0f), IMG_HF);
    float x2 = fminf(fmaxf(pcx + 0.5f * pw, 0.0f), IMG_WF);
    float y2 = fminf(fmaxf(pcy + 0.5f * ph, 0.0f), IMG_HF);
    *(float4*)&s_dec[i][0] = make_float4(x1, y1, x2, y2);
    bool valid = (best[e] >= FILT_T) && (bi[e] != NC - 1);
    s_key[i] = valid ? best[e] : -1.0f;     // matches -where(valid, smax, -1) order
    s_val[i] = (short)i;
    s_cls[i] = (unsigned char)bi[e];
  }
  __syncthreads();

  // ---- Phase 3: bitonic sort by key, descending (2048 elems, 1024 threads) ----
  for (int k = 2; k <= NN; k <<= 1) {
    for (int j = k >> 1; j > 0; j >>= 1) {
      #pragma unroll
      for (int e = 0; e < 2; ++e) {
        int i = tid + e * NTHREADS;
        int p = i ^ j;
        if (p > i) {
          float ki = s_key[i], kp = s_key[p];
          bool desc = ((i & k) == 0);
          if (desc ? (ki < kp) : (ki > kp)) {
            s_key[i] = kp; s_key[p] = ki;
            short vi = s_val[i]; s_val[i] = s_val[p]; s_val[p] = vi;
          }
        }
      }
      __syncthreads();
    }
  }

  // ---- Phase 4: kept-list greedy NMS on wave 0 (kept boxes live in registers,
  //      striped across the 32 lanes; suppression vote via wave32 ballot) ----
  if (tid < 32) {
    const int lane = tid;
    float4 kb0 = make_float4(0,0,0,0), kb1 = kb0, kb2 = kb0, kb3 = kb0;
    int kc0 = -1, kc1 = -1, kc2 = -1, kc3 = -1;
    int nkept = 0;

    for (int j = 0; j < NN && nkept < TOPK; ++j) {
      float sc = s_key[j];
      if (sc < 0.0f) break;               // invalid boxes sorted to the tail
      int ci = s_val[j];
      float4 cb = *(const float4*)&s_dec[ci][0];
      int ccls = s_cls[ci];

      bool sup = false;
      if ((lane +  0) < nkept && kc0 == ccls) sup = sup || (iou_f(cb, kb0) > NMS_T);
      if ((lane + 32) < nkept && kc1 == ccls) sup = sup || (iou_f(cb, kb1) > NMS_T);
      if ((lane + 64) < nkept && kc2 == ccls) sup = sup || (iou_f(cb, kb2) > NMS_T);
      if ((lane + 96) < nkept && kc3 == ccls) sup = sup || (iou_f(cb, kb3) > NMS_T);

      if (!__any((int)sup)) {
        int owner = nkept & 31;
        int slot  = nkept >> 5;
        if (lane == owner) {
          switch (slot) {
            case 0: kb0 = cb; kc0 = ccls; break;
            case 1: kb1 = cb; kc1 = ccls; break;
            case 2: kb2 = cb; kc2 = ccls; break;
            default: kb3 = cb; kc3 = ccls; break;
          }
        }
        if (lane == 0) {
          float* r = outB + (size_t)nkept * 6;
          r[0] = cb.x; r[1] = cb.y; r[2] = cb.z; r[3] = cb.w;
          r[4] = sc;   r[5] = (float)ccls;
        }
        ++nkept;
      }
    }

    // pad remaining rows with -1
    for (int r = nkept + lane; r < TOPK; r += 32) {
      float* o = outB + (size_t)r * 6;
      o[0] = o[1] = o[2] = o[3] = o[4] = o[5] = -1.0f;
    }
  }
}

extern "C" void kernel_launch(void* const* d_in, const int* in_sizes, int n_in,
                              void* d_out, int out_size, void* d_ws, size_t ws_size,
                              hipStream_t stream) {
  (void)in_sizes; (void)n_in; (void)out_size; (void)d_ws; (void)ws_size;
  const float* boxes  = (const float*)d_in[0];
  const float* scores = (const float*)d_in[1];
  const float* deltas = (const float*)d_in[2];
  float* out = (float*)d_out;
  rcnn_post_kernel<<<BB, NTHREADS, 0, stream>>>(boxes, scores, deltas, out);
}